// Context_CP_66460323938409
// MI455X (gfx1250) — compile-verified
//
#include <hip/hip_runtime.h>
#include <hip/hip_bf16.h>

typedef __attribute__((ext_vector_type(2))) float v2f;
typedef __attribute__((ext_vector_type(8))) float v8f;

#define B_ROWS   1024
#define RDIM     128
#define NB       50
#define VOCAB    100000

// ---------------------------------------------------------------------------
// Kernel A: gather + projection + masked attention + context; G = lhs*rel*e_c
// One block per batch row, 128 threads (4 wave32s).
// ---------------------------------------------------------------------------
__global__ __launch_bounds__(128)
void attn_kernel(const int* __restrict__ x,
                 const int* __restrict__ nb_idx,
                 const int* __restrict__ nb_len,
                 const float* __restrict__ E_lhs,
                 const float* __restrict__ E_rel,
                 const float* __restrict__ E_rhs,
                 const float* __restrict__ W_w,
                 const float* __restrict__ W_b,
                 float* __restrict__ G,
                 float* __restrict__ out_lhs,
                 float* __restrict__ out_rel,
                 float* __restrict__ out_rhs)
{
    const int b = blockIdx.x;
    const int t = threadIdx.x;          // 0..127
    const int lane = t & 31;
    const int wave = t >> 5;

    __shared__ float s_trp[3 * RDIM];   // [lhs | rel | rhs]
    __shared__ float s_w[RDIM];
    __shared__ float s_alpha[NB];

    const int i0 = x[b * 3 + 0];
    const int i1 = x[b * 3 + 1];
    const int i2 = x[b * 3 + 2];

    const float l = E_lhs[(long)i0 * RDIM + t];
    const float r = E_rel[(long)i1 * RDIM + t];
    const float q = E_rhs[(long)i2 * RDIM + t];
    s_trp[t]            = l;
    s_trp[RDIM + t]     = r;
    s_trp[2 * RDIM + t] = q;
    // Write-once outputs: non-temporal, don't pollute L2
    __builtin_nontemporal_store(l, out_lhs + b * RDIM + t);
    __builtin_nontemporal_store(r, out_rel + b * RDIM + t);
    __builtin_nontemporal_store(q, out_rhs + b * RDIM + t);
    __syncthreads();

    // w[t] = dot(trp_E, W_w row t) + W_b[t]   (W_w is (128, 384) row-major)
    float acc = W_b[t];
    const float* wr = W_w + (long)t * (3 * RDIM);
    #pragma unroll 8
    for (int i = 0; i < 3 * RDIM; ++i) acc += s_trp[i] * wr[i];
    s_w[t] = acc;
    __syncthreads();

    // logits: each wave handles neighbors m = wave, wave+4, ...
    const int len = nb_len[b];
    for (int m = wave; m < NB; m += 4) {
        const int v = nb_idx[b * NB + m];
        const float* er = E_rhs + (long)v * RDIM;
        float p = s_w[lane]      * er[lane]
                + s_w[lane + 32] * er[lane + 32]
                + s_w[lane + 64] * er[lane + 64]
                + s_w[lane + 96] * er[lane + 96];
        #pragma unroll
        for (int off = 16; off > 0; off >>= 1) p += __shfl_xor(p, off, 32);
        if (lane == 0) {
            // masked neighbors have nb_E == 0 -> logit exactly 0 (matches ref)
            s_alpha[m] = (m < len) ? p : 0.0f;
        }
    }
    __syncthreads();

    // softmax over all 50 entries (masked logits are 0, as in reference)
    if (t == 0) {
        float mx = -3.0e38f;
        for (int m = 0; m < NB; ++m) mx = fmaxf(mx, s_alpha[m]);
        float s = 0.0f;
        for (int m = 0; m < NB; ++m) {
            const float e = __expf(s_alpha[m] - mx);
            s_alpha[m] = e;
            s += e;
        }
        const float inv = 1.0f / s;
        for (int m = 0; m < NB; ++m) s_alpha[m] *= inv;
    }
    __syncthreads();

    // e_c[t] = sum over valid neighbors (masked rows contribute zero vectors)
    float ec = 0.0f;
    for (int m = 0; m < len; ++m) {
        const int v = nb_idx[b * NB + m];
        ec += s_alpha[m] * E_rhs[(long)v * RDIM + t];
    }
    G[b * RDIM + t] = l * r * ec;
}

// ---------------------------------------------------------------------------
// Kernel B: tot = G (1024x128) @ E_rhs^T (128x100000) via V_WMMA_F32_16X16X4_F32
// 256 threads = 8 waves; wave w of block (bx,by) computes M-tile (bx*8+w),
// N-strip by*80 .. by*80+79 (5 N-subtiles). 100000 = 1250*80, 1024 = 8*8*16.
// Memory-bound on the 409.6 MB output -> fp32 WMMA (no precision downgrade
// needed) + non-temporal stores so E_rhs (51.2 MB) stays L2-resident for its
// 64x reuse across M-tiles.
// ---------------------------------------------------------------------------
__global__ __launch_bounds__(256)
void gemm_wmma_kernel(const float* __restrict__ G,
                      const float* __restrict__ E_rhs,
                      float* __restrict__ out)
{
    const int lane = threadIdx.x & 31;
    const int wave = threadIdx.x >> 5;
    const int half = lane >> 4;         // 0: lanes 0-15, 1: lanes 16-31
    const int l16  = lane & 15;

    const int m0 = (blockIdx.x * 8 + wave) * 16;   // row tile base
    const int n0 = blockIdx.y * 80;                // col strip base

    v8f acc0 = {}, acc1 = {}, acc2 = {}, acc3 = {}, acc4 = {};

    // A frag: lane m = l16, VGPR pair holds K = k0+2*half, k0+2*half+1
    const float* ap = G + (long)(m0 + l16) * RDIM + half * 2;
    // B frag: lane n = l16 (per subtile), same K mapping; E_rhs row-major (V,128)
    const float* bp = E_rhs + (long)(n0 + l16) * RDIM + half * 2;

    #pragma unroll 4
    for (int k0 = 0; k0 < RDIM; k0 += 4) {
        const v2f a  = *(const v2f*)(ap + k0);
        const v2f b0 = *(const v2f*)(bp + k0);
        const v2f b1 = *(const v2f*)(bp + 16 * RDIM + k0);
        const v2f b2 = *(const v2f*)(bp + 32 * RDIM + k0);
        const v2f b3 = *(const v2f*)(bp + 48 * RDIM + k0);
        const v2f b4 = *(const v2f*)(bp + 64 * RDIM + k0);
        acc0 = __builtin_amdgcn_wmma_f32_16x16x4_f32(false, a, false, b0, (short)0, acc0, false, false);
        acc1 = __builtin_amdgcn_wmma_f32_16x16x4_f32(false, a, false, b1, (short)0, acc1, false, false);
        acc2 = __builtin_amdgcn_wmma_f32_16x16x4_f32(false, a, false, b2, (short)0, acc2, false, false);
        acc3 = __builtin_amdgcn_wmma_f32_16x16x4_f32(false, a, false, b3, (short)0, acc3, false, false);
        acc4 = __builtin_amdgcn_wmma_f32_16x16x4_f32(false, a, false, b4, (short)0, acc4, false, false);
    }

    // D layout: VGPR r -> row (r + 8*half), col = l16 (per 16-wide subtile).
    // 409.6 MB write-once stream: non-temporal so it bypasses L2 residency.
    #pragma unroll
    for (int r = 0; r < 8; ++r) {
        float* o = out + (long)(m0 + r + 8 * half) * VOCAB + n0 + l16;
        __builtin_nontemporal_store(acc0[r], o);
        __builtin_nontemporal_store(acc1[r], o + 16);
        __builtin_nontemporal_store(acc2[r], o + 32);
        __builtin_nontemporal_store(acc3[r], o + 48);
        __builtin_nontemporal_store(acc4[r], o + 64);
    }
}

extern "C" void kernel_launch(void* const* d_in, const int* in_sizes, int n_in,
                              void* d_out, int out_size, void* d_ws, size_t ws_size,
                              hipStream_t stream) {
    const int*   x      = (const int*)d_in[0];
    const int*   nb_idx = (const int*)d_in[1];
    const int*   nb_len = (const int*)d_in[2];
    const float* E_lhs  = (const float*)d_in[3];
    const float* E_rel  = (const float*)d_in[4];
    const float* E_rhs  = (const float*)d_in[5];
    const float* W_w    = (const float*)d_in[6];
    const float* W_b    = (const float*)d_in[7];

    float* out     = (float*)d_out;
    float* out_tot = out;                                   // 1024*100000
    float* out_lhs = out + (long)B_ROWS * VOCAB;            // 1024*128
    float* out_rel = out_lhs + B_ROWS * RDIM;
    float* out_rhs = out_rel + B_ROWS * RDIM;

    float* G = (float*)d_ws;                                // 1024*128 floats

    attn_kernel<<<B_ROWS, 128, 0, stream>>>(x, nb_idx, nb_len,
                                            E_lhs, E_rel, E_rhs,
                                            W_w, W_b, G,
                                            out_lhs, out_rel, out_rhs);

    dim3 grid(8, 1250);
    gemm_wmma_kernel<<<grid, 256, 0, stream>>>(G, E_rhs, out_tot);
}